// CustomGINEncoder_44813688767169
// MI455X (gfx1250) — compile-verified
//
#include <hip/hip_runtime.h>
#include <math.h>

// Problem constants (match reference)
#define NNODES 10000
#define NEDGES 160000
#define ETOT   (NEDGES + NNODES)   // 170000 edges incl. self loops
#define HEADS  6
#define DCH    128                 // channels per head / in-channels
#define HD     (HEADS * DCH)       // 768
#define TILES_N (HD / 16)          // 48
#define MBLK    5                  // 5 stacked 16-row tiles per wave = 80 rows
#define MBLK_ROWS (MBLK * 16)      // 80
#define MBLKS   (NNODES / MBLK_ROWS) // 125
#define NEG_SLOPE 0.2f

typedef float v2f __attribute__((ext_vector_type(2)));
typedef float v8f __attribute__((ext_vector_type(8)));

// ---- ordered-int encoding for float atomic max ----
__device__ __forceinline__ int f2ord(float f) {
    int i = __float_as_int(f);
    return (i >= 0) ? i : (i ^ 0x7FFFFFFF);
}
__device__ __forceinline__ float ord2f(int k) {
    return __int_as_float((k >= 0) ? k : (k ^ 0x7FFFFFFF));
}
#define ORD_NEG_INF ((int)0x807FFFFF)  // f2ord(-inf)

// ------------------------------------------------------------------
// XW = X[N,128] @ W[128,768] using V_WMMA_F32_16X16X4_F32.
// One wave computes an 80x16 block (5 accumulators) so each B fragment
// is reused 5x: per k-step 5xb64 (A) + 2xb32 (B) feed 5 WMMAs.
// Fragment layouts per CDNA5 ISA 7.12.2 (wave32):
//   A 16x4:  lanes 0-15 M=lane, VGPR0/1 = K={0,1}; lanes 16-31 K={2,3}
//   B 4x16:  lanes hold N=lane%16; VGPR0 = K {0|2 by half}, VGPR1 = K+1
//   C/D:     VGPR j -> row (j + 8*half), col = lane%16
// ------------------------------------------------------------------
__global__ __launch_bounds__(128)
void gat_gemm_wmma(const float* __restrict__ X, const float* __restrict__ W,
                   float* __restrict__ XW)
{
    const int wave = threadIdx.x >> 5;
    const int lane = threadIdx.x & 31;
    const int blk  = blockIdx.x * 4 + wave;          // < 6000, always valid
    const int tm   = (blk / TILES_N) * MBLK_ROWS;    // 0..9920
    const int tn   = (blk % TILES_N) * 16;           // 0..752
    const int half = lane >> 4;
    const int lm   = lane & 15;

    const float* __restrict__ arow = X + (size_t)(tm + lm) * DCH;
    const float* __restrict__ bcol = W + (tn + lm);

    v8f acc[MBLK];
#pragma unroll
    for (int t = 0; t < MBLK; ++t)
        acc[t] = (v8f){0.f, 0.f, 0.f, 0.f, 0.f, 0.f, 0.f, 0.f};

#pragma unroll 8
    for (int k0 = 0; k0 < DCH; k0 += 4) {
        const int ka = k0 + 2 * half;
        v2f b;
        b.x = bcol[(size_t)ka * HD];
        b.y = bcol[(size_t)(ka + 1) * HD];
#pragma unroll
        for (int t = 0; t < MBLK; ++t) {
            v2f a;
            a.x = arow[(size_t)t * 16 * DCH + ka];
            a.y = arow[(size_t)t * 16 * DCH + ka + 1];
            acc[t] = __builtin_amdgcn_wmma_f32_16x16x4_f32(
                /*neg_a=*/false, a, /*neg_b=*/false, b,
                /*c_mod=*/(short)0, acc[t], /*reuse_a=*/false, /*reuse_b=*/false);
        }
    }

#pragma unroll
    for (int t = 0; t < MBLK; ++t) {
        float* __restrict__ dst =
            XW + (size_t)(tm + t * 16 + 8 * half) * HD + tn + lm;
#pragma unroll
        for (int j = 0; j < 8; ++j)
            dst[(size_t)j * HD] = acc[t][j];
    }
}

// ------------------------------------------------------------------
// Per-(node,head): a_src = <xw, att_src>, a_dst = <xw, att_dst>.
// Also initializes the segment-max (ordered -inf) and denom (0).
// One wave32 per (n,h); 32 lanes x float4 covers 128 channels.
// ------------------------------------------------------------------
__global__ __launch_bounds__(256)
void gat_att_scores(const float* __restrict__ XW,
                    const float* __restrict__ att_src,
                    const float* __restrict__ att_dst,
                    float* __restrict__ a_src, float* __restrict__ a_dst,
                    int* __restrict__ mI, float* __restrict__ denom)
{
    const int warp = (blockIdx.x * blockDim.x + threadIdx.x) >> 5;
    const int lane = threadIdx.x & 31;
    if (warp >= NNODES * HEADS) return;
    const int n = warp / HEADS, h = warp % HEADS;

    const float4 xl = ((const float4*)(XW + (size_t)n * HD + h * DCH))[lane];
    const float4 al = ((const float4*)(att_src + h * DCH))[lane];
    const float4 dl = ((const float4*)(att_dst + h * DCH))[lane];

    float s = xl.x * al.x + xl.y * al.y + xl.z * al.z + xl.w * al.w;
    float d = xl.x * dl.x + xl.y * dl.y + xl.z * dl.z + xl.w * dl.w;
#pragma unroll
    for (int o = 16; o > 0; o >>= 1) {
        s += __shfl_xor(s, o, 32);
        d += __shfl_xor(d, o, 32);
    }
    if (lane == 0) {
        a_src[warp] = s;
        a_dst[warp] = d;
        mI[warp] = ORD_NEG_INF;
        denom[warp] = 0.f;
    }
}

__global__ void gat_zero_f32(float* __restrict__ p, int n)
{
    int i = blockIdx.x * blockDim.x + threadIdx.x;
    if (i < n) p[i] = 0.f;
}

__device__ __forceinline__ void edge_sd(const int* __restrict__ ei, int e,
                                        int& s, int& d)
{
    if (e < NEDGES) { s = ei[e]; d = ei[NEDGES + e]; }
    else            { s = d = e - NEDGES; }
}

// e = leaky_relu(a_src[src] + a_dst[dst]); segment max over dst (ordered-int atomics)
__global__ __launch_bounds__(256)
void gat_edge_logits(const int* __restrict__ ei,
                     const float* __restrict__ a_src,
                     const float* __restrict__ a_dst,
                     float* __restrict__ ebuf, int* __restrict__ mI)
{
    const int e = blockIdx.x * blockDim.x + threadIdx.x;
    if (e >= ETOT) return;
    int s, d; edge_sd(ei, e, s, d);
#pragma unroll
    for (int h = 0; h < HEADS; ++h) {
        float v = a_src[s * HEADS + h] + a_dst[d * HEADS + h];
        v = (v > 0.f) ? v : NEG_SLOPE * v;
        ebuf[(size_t)e * HEADS + h] = v;
        atomicMax(&mI[d * HEADS + h], f2ord(v));
    }
}

// e' = exp(e - m[dst]); segment sum into denom[dst]
__global__ __launch_bounds__(256)
void gat_edge_exp(const int* __restrict__ ei, float* __restrict__ ebuf,
                  const int* __restrict__ mI, float* __restrict__ denom)
{
    const int e = blockIdx.x * blockDim.x + threadIdx.x;
    if (e >= ETOT) return;
    int s, d; edge_sd(ei, e, s, d);
#pragma unroll
    for (int h = 0; h < HEADS; ++h) {
        const float m = ord2f(mI[d * HEADS + h]);
        const float v = __expf(ebuf[(size_t)e * HEADS + h] - m);
        ebuf[(size_t)e * HEADS + h] = v;
        atomicAdd(&denom[d * HEADS + h], v);
    }
}

// out[dst,h,:] += alpha(e,h) * xw[src,h,:]; one block per edge, L2-resident atomics
__global__ __launch_bounds__(256)
void gat_edge_scatter(const int* __restrict__ ei,
                      const float* __restrict__ ebuf,
                      const float* __restrict__ denom,
                      const float* __restrict__ XW, float* __restrict__ OUT)
{
    const int e = blockIdx.x;
    int s, d; edge_sd(ei, e, s, d);
    const float* __restrict__ xs = XW + (size_t)s * HD;
    float* __restrict__ od = OUT + (size_t)d * HD;
#pragma unroll
    for (int it = 0; it < 3; ++it) {
        const int c = it * 256 + threadIdx.x;       // 0..767
        const int h = c >> 7;
        const float alpha = ebuf[(size_t)e * HEADS + h] /
                            (denom[d * HEADS + h] + 1e-16f);
        atomicAdd(&od[c], alpha * xs[c]);
    }
}

// x_next[n,d] = relu(mean_h(out[n,h,d]) + bias[d])
__global__ __launch_bounds__(256)
void gat_mean_relu(const float* __restrict__ OUT, const float* __restrict__ bias,
                   float* __restrict__ Xn)
{
    const int i = blockIdx.x * blockDim.x + threadIdx.x;
    if (i >= NNODES * DCH) return;
    const int n = i >> 7, d = i & 127;
    const float* __restrict__ p = OUT + (size_t)n * HD + d;
    float s = 0.f;
#pragma unroll
    for (int h = 0; h < HEADS; ++h) s += p[h * DCH];
    s = s * (1.f / (float)HEADS) + bias[d];
    Xn[i] = (s > 0.f) ? s : 0.f;
}

// ------------------------------------------------------------------
extern "C" void kernel_launch(void* const* d_in, const int* in_sizes, int n_in,
                              void* d_out, int out_size, void* d_ws, size_t ws_size,
                              hipStream_t stream)
{
    const float* x  = (const float*)d_in[0];
    const int*   ei = (const int*)d_in[1];

    // Workspace carve-up (floats): ~77 MB total
    float* XW   = (float*)d_ws;                       // N*HD = 7,680,000
    float* OUT  = XW  + (size_t)NNODES * HD;          // N*HD = 7,680,000
    float* EBUF = OUT + (size_t)NNODES * HD;          // ETOT*H = 1,020,000
    float* ASRC = EBUF + (size_t)ETOT * HEADS;        // 60,000
    float* ADST = ASRC + (size_t)NNODES * HEADS;      // 60,000
    float* DEN  = ADST + (size_t)NNODES * HEADS;      // 60,000
    int*   MI   = (int*)(DEN + (size_t)NNODES * HEADS); // 60,000
    float* XA   = (float*)(MI + (size_t)NNODES * HEADS); // N*D = 1,280,000
    float* XB   = XA + (size_t)NNODES * DCH;             // N*D = 1,280,000

    const float* in_x = x;
    for (int layer = 0; layer < 4; ++layer) {
        const float* W  = (const float*)d_in[2 + 4 * layer];
        const float* as = (const float*)d_in[3 + 4 * layer];
        const float* ad = (const float*)d_in[4 + 4 * layer];
        const float* b  = (const float*)d_in[5 + 4 * layer];
        float* out_x = (layer == 3) ? (float*)d_out : ((layer & 1) ? XB : XA);

        // 1) XW = in_x @ W   (WMMA f32, 80x16 register blocks)
        gat_gemm_wmma<<<(MBLKS * TILES_N) / 4, 128, 0, stream>>>(in_x, W, XW);
        // 2) attention scores + init max/denom
        gat_att_scores<<<(NNODES * HEADS) / 8, 256, 0, stream>>>(
            XW, as, ad, ASRC, ADST, MI, DEN);
        // 3) zero the aggregation accumulator
        gat_zero_f32<<<(NNODES * HD + 255) / 256, 256, 0, stream>>>(OUT, NNODES * HD);
        // 4) edge logits + segment max
        gat_edge_logits<<<(ETOT + 255) / 256, 256, 0, stream>>>(ei, ASRC, ADST, EBUF, MI);
        // 5) exp + segment sum
        gat_edge_exp<<<(ETOT + 255) / 256, 256, 0, stream>>>(ei, EBUF, MI, DEN);
        // 6) weighted scatter-add
        gat_edge_scatter<<<ETOT, 256, 0, stream>>>(ei, EBUF, DEN, XW, OUT);
        // 7) head mean + bias + relu
        gat_mean_relu<<<(NNODES * DCH + 255) / 256, 256, 0, stream>>>(OUT, b, out_x);

        in_x = out_x;
    }
    (void)in_sizes; (void)n_in; (void)out_size; (void)ws_size;
}